// _ProposalLayer_27951647162816
// MI455X (gfx1250) — compile-verified
//
#include <hip/hip_runtime.h>
#include <stdint.h>

// CDNA5 / gfx1250: wave32, WMMA 16x16 shapes.
typedef __attribute__((ext_vector_type(2))) float v2f;
typedef __attribute__((ext_vector_type(8))) float v8f;
typedef __attribute__((ext_vector_type(2))) int   v2i;

#define AS_GLOBAL __attribute__((address_space(1)))
#define AS_LDS    __attribute__((address_space(3)))

namespace {
constexpr int   kA       = 10;
constexpr int   kL       = 4096;
constexpr int   kB       = 4;
constexpr int   kNProp   = kA * kL;   // 40960 proposals per image
constexpr int   kNSort   = 65536;     // power-of-two bitonic padding
constexpr int   kPre     = 6000;      // PRE_NMS_TOP_N (= 375 * 16 exactly)
constexpr int   kPost    = 300;       // POST_NMS_TOP_N
constexpr int   kNT      = 375;       // 16-wide tiles covering 6000
constexpr int   kSortPad = 6016;
constexpr float kThresh  = 0.7f;
}

__device__ __constant__ float c_scales[kA] =
    {2.f, 4.f, 5.f, 6.f, 8.f, 9.f, 10.f, 12.f, 14.f, 16.f};

// ---------------------------------------------------------------------------
// 1) Decode proposals + scores, build sortable 64-bit keys.
//    key = (order_preserving(score) << 32) | ~index  -> sort descending gives
//    descending score with ascending-index tie-break (matches stable argsort).
// ---------------------------------------------------------------------------
__global__ __launch_bounds__(256) void proposal_score_kernel(
    const float* __restrict__ scores_in, const float* __restrict__ twin_deltas,
    float2* __restrict__ prop, unsigned long long* __restrict__ keys)
{
  int t = blockIdx.x * 256 + threadIdx.x;      // kB * 65536 threads
  int b = t >> 16;
  int p = t & 0xFFFF;
  if (b >= kB) return;
  if (p >= kNProp) {                            // bitonic padding entries
    keys[t] = (unsigned long long)(unsigned)(~p);  // score key = 0 (below all)
    return;
  }
  int a = p % kA;
  int l = p / kA;
  float w = c_scales[a] * 8.f;                  // anchor width = scale*stride
  const float* tdb = twin_deltas + (size_t)b * (2 * kA) * kL;
  float d0 = tdb[(2 * a) * kL + l];
  float d1 = tdb[(2 * a + 1) * kL + l];
  float s  = scores_in[((size_t)b * (2 * kA) + kA + a) * kL + l];
  float ctr = 4.f + 8.f * (float)l;             // anchor center = 4 + 8l
  float pc  = fmaf(d0, w, ctr);
  float pl  = expf(d1) * w;
  float x1  = pc - 0.5f * pl;
  float x2  = pc + 0.5f * pl;
  x1 = fminf(fmaxf(x1, 0.f), 32767.f);          // clip to [0, L*8-1]
  x2 = fminf(fmaxf(x2, 0.f), 32767.f);
  if (x2 - x1 + 1.f < 8.f) s = 0.f;             // MIN_SIZE filter
  prop[(size_t)b * kNProp + p] = make_float2(x1, x2);
  unsigned u  = __float_as_uint(s);
  unsigned fs = (u & 0x80000000u) ? ~u : (u | 0x80000000u);
  keys[t] = ((unsigned long long)fs << 32) | (unsigned)(~p);
}

// ---------------------------------------------------------------------------
// 2) Bitonic sort (descending), 65536 keys per image, image-local indexing.
//    Tiles are staged global->LDS with GLOBAL_LOAD_ASYNC_TO_LDS_B64 (ASYNCcnt),
//    the CDNA5 bulk LDS-fill path that bypasses VGPRs.
// ---------------------------------------------------------------------------
__global__ __launch_bounds__(1024) void sort_local_full(unsigned long long* __restrict__ keys)
{
  __shared__ unsigned long long sk[2048];
  int tid = threadIdx.x;
  size_t base = (size_t)blockIdx.x * 2048;
  int lbase = (blockIdx.x & 31) * 2048;         // image-local tile base
  __builtin_amdgcn_global_load_async_to_lds_b64(
      (AS_GLOBAL v2i*)(keys + base + tid),
      (AS_LDS v2i*)&sk[tid], 0, 0);
  __builtin_amdgcn_global_load_async_to_lds_b64(
      (AS_GLOBAL v2i*)(keys + base + tid + 1024),
      (AS_LDS v2i*)&sk[tid + 1024], 0, 0);
  asm volatile("s_wait_asynccnt 0x0" ::: "memory");
  __syncthreads();
  for (int k = 2; k <= 2048; k <<= 1) {
    for (int j = k >> 1; j >= 1; j >>= 1) {
      int i = ((tid & ~(j - 1)) << 1) | (tid & (j - 1));
      bool up = (((lbase + i) & k) == 0);
      unsigned long long a = sk[i], c = sk[i + j];
      if ((a < c) == up) { sk[i] = c; sk[i + j] = a; }
      __syncthreads();
    }
  }
  keys[base + tid]        = sk[tid];
  keys[base + tid + 1024] = sk[tid + 1024];
}

__global__ __launch_bounds__(256) void sort_global_pass(
    unsigned long long* __restrict__ keys, int k, int j)
{
  int t = blockIdx.x * 256 + threadIdx.x;       // kB * 32768 threads
  int b = t >> 15;
  if (b >= kB) return;
  int tt = t & 32767;
  int i = ((tt & ~(j - 1)) << 1) | (tt & (j - 1));
  size_t o = (size_t)b << 16;
  bool up = ((i & k) == 0);
  unsigned long long a = keys[o + i], c = keys[o + i + j];
  if ((a < c) == up) { keys[o + i] = c; keys[o + i + j] = a; }
}

__global__ __launch_bounds__(1024) void sort_local_merge(
    unsigned long long* __restrict__ keys, int k)
{
  __shared__ unsigned long long sk[2048];
  int tid = threadIdx.x;
  size_t base = (size_t)blockIdx.x * 2048;
  int lbase = (blockIdx.x & 31) * 2048;
  __builtin_amdgcn_global_load_async_to_lds_b64(
      (AS_GLOBAL v2i*)(keys + base + tid),
      (AS_LDS v2i*)&sk[tid], 0, 0);
  __builtin_amdgcn_global_load_async_to_lds_b64(
      (AS_GLOBAL v2i*)(keys + base + tid + 1024),
      (AS_LDS v2i*)&sk[tid + 1024], 0, 0);
  asm volatile("s_wait_asynccnt 0x0" ::: "memory");
  __syncthreads();
  bool up = ((lbase & k) == 0);                 // k >= 4096 > tile span: uniform
  for (int j = 1024; j >= 1; j >>= 1) {
    int i = ((tid & ~(j - 1)) << 1) | (tid & (j - 1));
    unsigned long long a = sk[i], c = sk[i + j];
    if ((a < c) == up) { sk[i] = c; sk[i + j] = a; }
    __syncthreads();
  }
  keys[base + tid]        = sk[tid];
  keys[base + tid + 1024] = sk[tid + 1024];
}

// ---------------------------------------------------------------------------
// 3) Gather top-6000 sorted proposals per image.
// ---------------------------------------------------------------------------
__global__ __launch_bounds__(256) void gather_sorted(
    const unsigned long long* __restrict__ keys, const float2* __restrict__ prop,
    float2* __restrict__ propS)
{
  int t = blockIdx.x * 256 + threadIdx.x;
  if (t >= kB * kSortPad) return;
  int b = t / kSortPad;
  int r = t - b * kSortPad;
  float2 v = make_float2(0.f, 0.f);
  if (r < kPre) {
    unsigned idx = ~(unsigned)(keys[((size_t)b << 16) + r] & 0xFFFFFFFFull);
    idx &= 0xFFFFu;
    if (idx < (unsigned)kNProp) v = prop[(size_t)b * kNProp + idx];
  }
  propS[(size_t)b * kSortPad + r] = v;
}

// ---------------------------------------------------------------------------
// 4) NMS suppression bitmask via WMMA.
//    One wave per 16x16 tile. Rank-2 pairwise matrices on the matrix pipe:
//      S1 = x1_i + x1_j, D1 = x1_i - x1_j, S2 = x2_i + x2_j, D2 = x2_i - x2_j
//    each as V_WMMA_F32_16X16X4_F32 (A row = (x, +/-1, 0, 0), B = (1; x_j)).
//    Then: min(x2) = 0.5(S2-|D2|), max(x1) = 0.5(S1+|D1|),
//          inter = max(0, min-max+1), len_i+len_j = S2-S1+2,
//          iou>t  <=>  inter*(1+t) > t*(len_i+len_j)  (division-free).
//    Mask is tile-major: mask[b][ti][tj][16 row-words] so lanes 0..15 emit one
//    coalesced 32B store per tile (ballot results are wave-uniform).
// ---------------------------------------------------------------------------
__global__ __launch_bounds__(256) void nms_mask_wmma(
    const float2* __restrict__ propS, unsigned short* __restrict__ mask)
{
  int wave = threadIdx.x >> 5;
  int lane = threadIdx.x & 31;
  int gw = blockIdx.x * 8 + wave;
  int total = kB * kNT * kNT;
  if (gw >= total) return;                      // uniform per wave
  int b  = gw / (kNT * kNT);
  int r  = gw - b * (kNT * kNT);
  int ti = r / kNT;
  int tj = r - ti * kNT;
  unsigned short* mtile = mask + ((size_t)(b * kNT + ti) * kNT + tj) * 16;
  if (tj < ti) {                                // j>i only: lower tiles all-zero
    if (lane < 16) mtile[lane] = 0;             // coalesced 32B zero-fill
    return;
  }
  const float2* ps = propS + (size_t)b * kSortPad;
  float2 rb = ps[ti * 16 + (lane & 15)];        // row boxes (i)
  float2 cb = ps[tj * 16 + (lane & 15)];        // col boxes (j)
  __builtin_prefetch(&ps[((tj + 1) % kNT) * 16], 0, 1);  // global_prefetch_b8

  // A 16x4 f32: lanes 0-15 hold (K0,K1) for row M=lane; lanes 16-31 hold K2,K3
  // (zeroed). B 4x16 f32: row0 = ones, row1 = x_j; K2/K3 rows zeroed.
  bool lo = lane < 16;
  float rx1 = lo ? rb.x : 0.f;
  float rx2 = lo ? rb.y : 0.f;
  float one = lo ? 1.f : 0.f;
  v2f aS1 = {rx1,  one};
  v2f aD1 = {rx1, -one};
  v2f aS2 = {rx2,  one};
  v2f aD2 = {rx2, -one};
  v2f b1  = {one, lo ? cb.x : 0.f};
  v2f b2  = {one, lo ? cb.y : 0.f};
  v8f z = {};
  v8f S1 = __builtin_amdgcn_wmma_f32_16x16x4_f32(false, aS1, false, b1, (short)0, z, false, false);
  v8f D1 = __builtin_amdgcn_wmma_f32_16x16x4_f32(false, aD1, false, b1, (short)0, z, false, false);
  v8f S2 = __builtin_amdgcn_wmma_f32_16x16x4_f32(false, aS2, false, b2, (short)0, z, false, false);
  v8f D2 = __builtin_amdgcn_wmma_f32_16x16x4_f32(false, aD2, false, b2, (short)0, z, false, false);

  int mhi = (lane >= 16) ? 8 : 0;               // C/D layout: hi lanes = rows M+8
  int jg  = tj * 16 + (lane & 15);
  unsigned bals[8];
#pragma unroll
  for (int v = 0; v < 8; ++v) {
    int ig = ti * 16 + v + mhi;
    float minx2  = 0.5f * (S2[v] - fabsf(D2[v]));
    float maxx1  = 0.5f * (S1[v] + fabsf(D1[v]));
    float inter  = fmaxf(0.f, minx2 - maxx1 + 1.f);
    float lensum = S2[v] - S1[v] + 2.f;
    bool sup = (inter * (1.f + kThresh) > kThresh * lensum) && (jg > ig);
    bals[v] = __builtin_amdgcn_ballot_w32(sup); // uniform: bits0-15 row v, 16-31 row v+8
  }
  if (lane < 16) {                              // single coalesced 32B store
    int vv = lane & 7;
    unsigned bb = (vv < 4) ? ((vv < 2) ? (vv == 0 ? bals[0] : bals[1])
                                       : (vv == 2 ? bals[2] : bals[3]))
                           : ((vv < 6) ? (vv == 4 ? bals[4] : bals[5])
                                       : (vv == 6 ? bals[6] : bals[7]));
    unsigned short val = (lane < 8) ? (unsigned short)(bb & 0xFFFFu)
                                    : (unsigned short)(bb >> 16);
    mtile[lane] = val;                          // row r = lane of this tile
  }
}

// ---------------------------------------------------------------------------
// 5) Sequential greedy scan: one wave32 per image, remv bitset in LDS.
//    Wave-synchronous (per-wave LDS is in-order); explicit s_wait_dscnt pins it.
// ---------------------------------------------------------------------------
__global__ __launch_bounds__(128) void nms_scan(
    const unsigned short* __restrict__ mask, int* __restrict__ keeplist,
    int* __restrict__ kcnt)
{
  __shared__ unsigned short remv_s[kB][kNT + 9];
  volatile unsigned short (*remv)[kNT + 9] = remv_s;
  int b    = threadIdx.x >> 5;                  // wave -> image
  int lane = threadIdx.x & 31;
  for (int t = lane; t < kNT; t += 32) remv[b][t] = 0;
  asm volatile("s_wait_dscnt 0x0" ::: "memory");
  const unsigned short* mimg = mask + (size_t)b * kNT * kNT * 16;
  int kept = 0;
  for (int i = 0; i < kPre; ++i) {
    unsigned w = remv[b][i >> 4];               // LDS broadcast to all lanes
    if (!((w >> (i & 15)) & 1u)) {
      if (lane == 0) keeplist[b * kPost + kept] = i;
      ++kept;
      if (kept >= kPost) break;                 // only first 300 kept matter
      // row i bits for column-tile t live at tile (i>>4, t), word (i&15)
      const unsigned short* rowb = mimg + ((size_t)(i >> 4) * kNT) * 16 + (i & 15);
      for (int t = lane; t < kNT; t += 32)
        remv[b][t] = (unsigned short)(remv[b][t] | rowb[(size_t)t * 16]);
      asm volatile("s_wait_dscnt 0x0" ::: "memory");
    }
  }
  if (lane == 0) kcnt[b] = kept;
}

// ---------------------------------------------------------------------------
// 6) Emit (batch_id, x1, x2), zero-padded past kept count.
// ---------------------------------------------------------------------------
__global__ __launch_bounds__(128) void write_output(
    const float2* __restrict__ propS, const int* __restrict__ keeplist,
    const int* __restrict__ kcnt, float* __restrict__ out)
{
  int t = blockIdx.x * 128 + threadIdx.x;
  if (t >= kB * kPost) return;
  int b = t / kPost;
  int r = t - b * kPost;
  float x1 = 0.f, x2 = 0.f;
  if (r < kcnt[b]) {
    int i = keeplist[b * kPost + r];
    float2 p = propS[(size_t)b * kSortPad + i];
    x1 = p.x; x2 = p.y;
  }
  out[(size_t)t * 3 + 0] = (float)b;
  out[(size_t)t * 3 + 1] = x1;
  out[(size_t)t * 3 + 2] = x2;
}

extern "C" void kernel_launch(void* const* d_in, const int* in_sizes, int n_in,
                              void* d_out, int out_size, void* d_ws, size_t ws_size,
                              hipStream_t stream)
{
  (void)in_sizes; (void)n_in; (void)out_size; (void)ws_size;
  const float* scores_in   = (const float*)d_in[0];
  const float* twin_deltas = (const float*)d_in[1];
  float* out = (float*)d_out;

  char* ws = (char*)d_ws;
  size_t off = 0;
  auto carve = [&](size_t bytes) {
    size_t o = off;
    off += (bytes + 255) & ~(size_t)255;
    return o;
  };
  unsigned long long* keys = (unsigned long long*)(ws + carve((size_t)kB * kNSort * 8));
  float2* prop             = (float2*)(ws + carve((size_t)kB * kNProp * sizeof(float2)));
  float2* propS            = (float2*)(ws + carve((size_t)kB * kSortPad * sizeof(float2)));
  unsigned short* mask     = (unsigned short*)(ws + carve((size_t)kB * kNT * kNT * 16 * 2));
  int* keeplist            = (int*)(ws + carve((size_t)kB * kPost * 4));
  int* kcnt                = (int*)(ws + carve((size_t)kB * 4));

  // 1) decode + keys
  proposal_score_kernel<<<(kB * kNSort) / 256, 256, 0, stream>>>(
      scores_in, twin_deltas, prop, keys);
  // 2) bitonic sort, descending, per image
  sort_local_full<<<kB * 32, 1024, 0, stream>>>(keys);
  for (int k = 4096; k <= kNSort; k <<= 1) {
    for (int j = k >> 1; j >= 2048; j >>= 1)
      sort_global_pass<<<(kB * kNSort / 2) / 256, 256, 0, stream>>>(keys, k, j);
    sort_local_merge<<<kB * 32, 1024, 0, stream>>>(keys, k);
  }
  // 3) gather top-6000
  gather_sorted<<<(kB * kSortPad + 255) / 256, 256, 0, stream>>>(keys, prop, propS);
  // 4) WMMA suppression bitmask
  int totalWaves = kB * kNT * kNT;
  nms_mask_wmma<<<(totalWaves + 7) / 8, 256, 0, stream>>>(propS, mask);
  // 5) greedy scan (1 wave per image)
  nms_scan<<<1, 128, 0, stream>>>(mask, keeplist, kcnt);
  // 6) output
  write_output<<<(kB * kPost + 127) / 128, 128, 0, stream>>>(propS, keeplist, kcnt, out);
}